// RelationNet_18889266168147
// MI455X (gfx1250) — compile-verified
//
#include <hip/hip_runtime.h>

// ---------------------------------------------------------------------------
// RelationNet forward for MI455X (gfx1250, wave32).
// f32 end-to-end; WMMA f32 16x16x4 for the GEMM tail; deterministic BN
// reductions (per-block partials, no float atomics).
// ---------------------------------------------------------------------------

typedef __attribute__((ext_vector_type(2))) float v2f;
typedef __attribute__((ext_vector_type(8))) float v8f;

__device__ __forceinline__ float wred32(float v) {
  #pragma unroll
  for (int o = 16; o > 0; o >>= 1) v += __shfl_down(v, o, 32);
  return v;
}
__device__ __forceinline__ float wred16(float v) {
  #pragma unroll
  for (int o = 8; o > 0; o >>= 1) v += __shfl_down(v, o, 16);
  return v;
}

__device__ __forceinline__ v8f wmma4(v2f a, v2f b, v8f c) {
  // D = A(16x4,f32) * B(4x16,f32) + C(16x16,f32)
  return __builtin_amdgcn_wmma_f32_16x16x4_f32(
      /*neg_a=*/false, a, /*neg_b=*/false, b,
      /*c_mod=*/(short)0, c, /*reuse_a=*/false, /*reuse_b=*/false);
}

// -------------------- conv1: (32,3,64,64) -> (32,24,32,32) ------------------
// grid (32 rows, 32 batch), block 768 = 24 waves; wave w == output channel w.
__global__ void k_conv1(const float* __restrict__ img,
                        const float* __restrict__ w,
                        const float* __restrict__ b,
                        float* __restrict__ y1,
                        float* __restrict__ part1 /* [2][24][1024] */) {
  __shared__ float xs[3 * 3 * 66];   // c x kh x (w halo)
  __shared__ float wl[24 * 27];
  const int i = blockIdx.x, n = blockIdx.y;
  const int tid = threadIdx.x;
  const int o = tid >> 5, j = tid & 31;
  if (tid < 594) {
    const int c = tid / 198, rem = tid % 198, r = rem / 66, wc = rem % 66;
    const int h = 2 * i - 1 + r, gw = wc - 1;
    float v = 0.0f;
    if ((unsigned)h < 64u && (unsigned)gw < 64u)
      v = img[((n * 3 + c) * 64 + h) * 64 + gw];
    xs[tid] = v;
  }
  if (tid < 648) wl[tid] = w[tid];
  __syncthreads();
  float acc = b[o];
  #pragma unroll
  for (int c = 0; c < 3; ++c)
    #pragma unroll
    for (int kh = 0; kh < 3; ++kh)
      #pragma unroll
      for (int kw = 0; kw < 3; ++kw)
        acc += xs[(c * 3 + kh) * 66 + 2 * j + kw] * wl[o * 27 + c * 9 + kh * 3 + kw];
  y1[((n * 24 + o) * 32 + i) * 32 + j] = acc;
  const float s1 = wred32(acc);
  const float s2 = wred32(acc * acc);
  if ((tid & 31) == 0) {
    const int blk = n * 32 + i;                  // 0..1023
    part1[o * 1024 + blk] = s1;
    part1[(24 + o) * 1024 + blk] = s2;
  }
}

// -------------------- BN finalize: partials -> scale/shift ------------------
// grid 24 (channel), block 256.
__global__ void k_bnstats(const float* __restrict__ part, int nblk,
                          const float* __restrict__ g,
                          const float* __restrict__ bt,
                          float* __restrict__ scale,
                          float* __restrict__ shift, float inv_count) {
  __shared__ float sm[16];
  const int c = blockIdx.x;
  const int tid = threadIdx.x;
  float s = 0.0f, q = 0.0f;
  for (int t = tid; t < nblk; t += 256) {
    s += part[c * nblk + t];
    q += part[(24 + c) * nblk + t];
  }
  s = wred32(s);
  q = wred32(q);
  const int wid = tid >> 5, lane = tid & 31;
  if (lane == 0) { sm[wid] = s; sm[8 + wid] = q; }
  __syncthreads();
  if (tid == 0) {
    float S = 0.0f, Q = 0.0f;
    #pragma unroll
    for (int k = 0; k < 8; ++k) { S += sm[k]; Q += sm[8 + k]; }
    const float mu  = S * inv_count;
    const float var = Q * inv_count - mu * mu;
    const float inv = rsqrtf(var + 1e-5f);
    const float sc  = g[c] * inv;
    scale[c] = sc;
    shift[c] = bt[c] - mu * sc;
  }
}

// -------------------- conv2: (32,24,32,32) -> (32,24,16,16) -----------------
// BN1 affine + ReLU folded into the LDS staging of the input tile.
// grid (16 rows, 32 batch), block 384; each 16-lane group == one out channel.
__global__ void k_conv2(const float* __restrict__ y1,
                        const float* __restrict__ w,
                        const float* __restrict__ b,
                        const float* __restrict__ scale1,
                        const float* __restrict__ shift1,
                        float* __restrict__ y2,
                        float* __restrict__ part2 /* [2][24][512] */) {
  __shared__ float a1[24 * 3 * 34];  // 2448 floats
  __shared__ float wl[24 * 24 * 9];  // 5184 floats
  const int i = blockIdx.x, n = blockIdx.y;
  const int tid = threadIdx.x;
  const int o = tid >> 4, j = tid & 15;
  for (int t = tid; t < 2448; t += 384) {
    const int c = t / 102, rem = t % 102, r = rem / 34, wc = rem % 34;
    const int h = 2 * i - 1 + r, gw = wc - 1;
    float v = 0.0f;
    if ((unsigned)h < 32u && (unsigned)gw < 32u) {
      const float raw = y1[((n * 24 + c) * 32 + h) * 32 + gw];
      v = raw * scale1[c] + shift1[c];
      v = v > 0.0f ? v : 0.0f;
    }
    a1[t] = v;
  }
  for (int t = tid; t < 5184; t += 384) wl[t] = w[t];
  __syncthreads();
  float acc = b[o];
  for (int c = 0; c < 24; ++c) {
    const float* ap = &a1[c * 102];
    const float* wp = &wl[o * 216 + c * 9];
    #pragma unroll
    for (int kh = 0; kh < 3; ++kh)
      #pragma unroll
      for (int kw = 0; kw < 3; ++kw)
        acc += ap[kh * 34 + 2 * j + kw] * wp[kh * 3 + kw];
  }
  y2[((n * 24 + o) * 16 + i) * 16 + j] = acc;
  const float s1 = wred16(acc);
  const float s2 = wred16(acc * acc);
  if ((tid & 15) == 0) {
    const int blk = n * 16 + i;                  // 0..511
    part2[o * 512 + blk] = s1;
    part2[(24 + o) * 512 + blk] = s2;
  }
}

// -------- per-sample feature sums s[n][c] = sum_hw relu(bn2(y2)) ------------
// grid 32 (batch), block 256 (one lane per spatial position).
__global__ void k_featsum(const float* __restrict__ y2,
                          const float* __restrict__ scale2,
                          const float* __restrict__ shift2,
                          float* __restrict__ sfeat /* [32][26] */) {
  __shared__ float part[8];
  const int n = blockIdx.x, tid = threadIdx.x;
  const int wid = tid >> 5, lane = tid & 31;
  for (int c = 0; c < 24; ++c) {
    float v = y2[(n * 24 + c) * 256 + tid] * scale2[c] + shift2[c];
    v = v > 0.0f ? v : 0.0f;
    v = wred32(v);
    if (lane == 0) part[wid] = v;
    __syncthreads();
    if (tid == 0) {
      float t = 0.0f;
      #pragma unroll
      for (int k = 0; k < 8; ++k) t += part[k];
      sfeat[n * 26 + c] = t;
    }
    __syncthreads();
  }
  // coordinate channels: linspace(-1,1) sums are (numerically) zero
  if (tid == 0) { sfeat[n * 26 + 24] = 0.0f; sfeat[n * 26 + 25] = 0.0f; }
}

// -------------------- relations (WMMA f32 16x16x4) --------------------------
// rel = (256*s) @ (Wi+Wj)  +  (65536*ques) @ Wq  +  65536*b_rel
// 1 block, 512 threads = 16 waves = 16 output tiles (2 M x 8 N).
__global__ void k_relations(const float* __restrict__ ques,
                            const float* __restrict__ w_rel,
                            const float* __restrict__ b_rel,
                            const float* __restrict__ sfeat,
                            float* __restrict__ rel /* [32][128] */) {
  const int lane = threadIdx.x & 31;
  const int wv = threadIdx.x >> 5;     // 0..15
  const int mt = wv >> 3, nt = wv & 7;
  const int half = lane >> 4, lsub = lane & 15;
  const int mrow = mt * 16 + lsub;     // A-matrix row held by this lane
  const int ncol = nt * 16 + lsub;     // B/C column held by this lane
  v8f c = {};
  // (65536*ques)(32x128) @ Wq(128x128) ; Wq = w_rel rows 52..179
  for (int k0 = 0; k0 < 128; k0 += 4) {
    const int ks = k0 + 2 * half;
    v2f a, b;
    a.x = ques[mrow * 128 + ks]     * 65536.0f;
    a.y = ques[mrow * 128 + ks + 1] * 65536.0f;
    b.x = w_rel[(52 + ks)     * 128 + ncol];
    b.y = w_rel[(52 + ks + 1) * 128 + ncol];
    c = wmma4(a, b, c);
  }
  // (256*s)(32x26) @ (Wi+Wj)(26x128), K padded to 28 with zeros
  for (int k0 = 0; k0 < 28; k0 += 4) {
    const int ks = k0 + 2 * half;
    v2f a = {}, b = {};
    if (ks < 26) {
      a.x = sfeat[mrow * 26 + ks] * 256.0f;
      b.x = w_rel[ks * 128 + ncol] + w_rel[(26 + ks) * 128 + ncol];
    }
    if (ks + 1 < 26) {
      a.y = sfeat[mrow * 26 + ks + 1] * 256.0f;
      b.y = w_rel[(ks + 1) * 128 + ncol] + w_rel[(26 + ks + 1) * 128 + ncol];
    }
    c = wmma4(a, b, c);
  }
  const float bias = b_rel[ncol] * 65536.0f;
  #pragma unroll
  for (int i = 0; i < 8; ++i) {
    const int m = mt * 16 + half * 8 + i;
    rel[m * 128 + ncol] = c[i] + bias;
  }
}

// -------------------- fc1 (WMMA f32 16x16x4) + bias + ReLU ------------------
// (32x128) @ (128x1024) ; grid 8 x 512 threads = 128 waves = 128 tiles.
__global__ void k_fc1(const float* __restrict__ rel,
                      const float* __restrict__ w1,
                      const float* __restrict__ b1,
                      float* __restrict__ h /* [32][1024] */) {
  const int gw = blockIdx.x * (blockDim.x >> 5) + (threadIdx.x >> 5); // 0..127
  const int lane = threadIdx.x & 31;
  const int mt = gw >> 6, nt = gw & 63;
  const int half = lane >> 4, lsub = lane & 15;
  const int mrow = mt * 16 + lsub;
  const int ncol = nt * 16 + lsub;
  v8f c = {};
  for (int k0 = 0; k0 < 128; k0 += 4) {
    const int ks = k0 + 2 * half;
    v2f a, b;
    a.x = rel[mrow * 128 + ks];
    a.y = rel[mrow * 128 + ks + 1];
    b.x = w1[ks * 1024 + ncol];
    b.y = w1[(ks + 1) * 1024 + ncol];
    if (k0 < 96) __builtin_prefetch(&w1[(ks + 8) * 1024 + ncol], 0, 1);
    c = wmma4(a, b, c);
  }
  const float bias = b1[ncol];
  #pragma unroll
  for (int i = 0; i < 8; ++i) {
    const int m = mt * 16 + half * 8 + i;
    const float v = c[i] + bias;
    h[m * 1024 + ncol] = v > 0.0f ? v : 0.0f;
  }
}

// -------------------- fc2: (32x1024) @ (1024x2) -----------------------------
__global__ void k_fc2(const float* __restrict__ h,
                      const float* __restrict__ w2,
                      const float* __restrict__ b2,
                      float* __restrict__ out /* [32][2] */) {
  const int t = threadIdx.x;  // 64 threads
  if (t < 64) {
    const int m = t >> 1, jj = t & 1;
    float acc = b2[jj];
    for (int k = 0; k < 1024; ++k) acc += h[m * 1024 + k] * w2[k * 2 + jj];
    out[m * 2 + jj] = acc;
  }
}

// ---------------------------------------------------------------------------
extern "C" void kernel_launch(void* const* d_in, const int* in_sizes, int n_in,
                              void* d_out, int out_size, void* d_ws, size_t ws_size,
                              hipStream_t stream) {
  (void)in_sizes; (void)n_in; (void)out_size; (void)ws_size;
  const float* image   = (const float*)d_in[0];
  const float* ques    = (const float*)d_in[1];
  const float* conv1_w = (const float*)d_in[2];
  const float* conv1_b = (const float*)d_in[3];
  const float* bn1_g   = (const float*)d_in[4];
  const float* bn1_b   = (const float*)d_in[5];
  const float* conv2_w = (const float*)d_in[6];
  const float* conv2_b = (const float*)d_in[7];
  const float* bn2_g   = (const float*)d_in[8];
  const float* bn2_b   = (const float*)d_in[9];
  const float* w_rel   = (const float*)d_in[10];
  const float* b_rel   = (const float*)d_in[11];
  const float* w_fc1   = (const float*)d_in[12];
  const float* b_fc1   = (const float*)d_in[13];
  const float* w_fc2   = (const float*)d_in[14];
  const float* b_fc2   = (const float*)d_in[15];

  float* ws = (float*)d_ws;
  // workspace layout (floats)
  float* y1     = ws;                    // 786432
  float* y2     = ws + 786432;           // 196608
  float* part1  = ws + 983040;           // 48*1024 = 49152
  float* part2  = ws + 1032192;          // 48*512  = 24576
  float* scale1 = ws + 1056768;          // 24
  float* shift1 = ws + 1056792;          // 24
  float* scale2 = ws + 1056816;          // 24
  float* shift2 = ws + 1056840;          // 24
  float* sfeat  = ws + 1056864;          // 32*26 = 832
  float* rel    = ws + 1057696;          // 32*128 = 4096
  float* hbuf   = ws + 1061792;          // 32*1024 = 32768

  k_conv1<<<dim3(32, 32), 768, 0, stream>>>(image, conv1_w, conv1_b, y1, part1);
  k_bnstats<<<24, 256, 0, stream>>>(part1, 1024, bn1_g, bn1_b, scale1, shift1,
                                    1.0f / 32768.0f);
  k_conv2<<<dim3(16, 32), 384, 0, stream>>>(y1, conv2_w, conv2_b, scale1, shift1,
                                            y2, part2);
  k_bnstats<<<24, 256, 0, stream>>>(part2, 512, bn2_g, bn2_b, scale2, shift2,
                                    1.0f / 8192.0f);
  k_featsum<<<32, 256, 0, stream>>>(y2, scale2, shift2, sfeat);
  k_relations<<<1, 512, 0, stream>>>(ques, w_rel, b_rel, sfeat, rel);
  k_fc1<<<8, 512, 0, stream>>>(rel, w_fc1, b_fc1, hbuf);
  k_fc2<<<1, 64, 0, stream>>>(hbuf, w_fc2, b_fc2, (float*)d_out);
}